// NeighborhoodDeviationLoss_80891414052988
// MI455X (gfx1250) — compile-verified
//
#include <hip/hip_runtime.h>
#include <hip/hip_bf16.h>

typedef __attribute__((ext_vector_type(2))) float v2f;
typedef __attribute__((ext_vector_type(8))) float v8f;

#define BATCH 4096
#define DIM   512
#define KNEI  5
#define WAVES 8
#define NPART 1024
#define COLTILES (BATCH / 16)   // 256 column tiles of 16

// ---------------------------------------------------------------------------
// Kernel 1: sq[i] = sum_d mean1[i][d]^2        (one 128-thread block per row)
// ---------------------------------------------------------------------------
__global__ __launch_bounds__(128)
void ndl_rownorm(const float* __restrict__ mean1, float* __restrict__ sq) {
    __shared__ float red[128];
    const int row = blockIdx.x;
    const int tid = threadIdx.x;
    float s = 0.0f;
    for (int d = tid; d < DIM; d += 128) {
        float v = mean1[(size_t)row * DIM + d];
        s += v * v;
    }
    red[tid] = s;
    __syncthreads();
    for (int st = 64; st > 0; st >>= 1) {
        if (tid < st) red[tid] += red[tid + st];
        __syncthreads();
    }
    if (tid == 0) sq[row] = red[0];
}

// ---------------------------------------------------------------------------
// Kernel 2: fused WMMA GEMM (mean1 @ mean1^T) -> squared distances -> top-5.
// Block = 256 threads (8 waves), handles 16 rows; each wave strides over
// 16-column tiles. fp32 WMMA: V_WMMA_F32_16X16X4_F32, K accumulated 4/step.
// NOTE: sqrt is monotone, so top-5 selection uses clamped SQUARED distances
// (identical indices, no per-tile sqrt fixup sequence in the hot loop).
// ---------------------------------------------------------------------------
__global__ __launch_bounds__(256)
void ndl_dist_topk(const float* __restrict__ mean1,
                   const float* __restrict__ sq,
                   int* __restrict__ topk_idx) {
    __shared__ float shA[16 * DIM];                 // 32 KB: this block's 16 rows
    __shared__ float shTile[WAVES][16][17];         // d2 tiles (pad 17 vs bank conflicts)
    __shared__ float shVal[16][WAVES][KNEI];        // per-wave top-5 values (d2)
    __shared__ int   shIdx[16][WAVES][KNEI];        // per-wave top-5 indices
    __shared__ float shSqRow[16];

    const int tid  = threadIdx.x;
    const int wave = tid >> 5;
    const int lane = tid & 31;
    const int r0   = blockIdx.x * 16;

    // Cooperative stage of A rows (16 x 512 f32) into LDS.
    const float4* gsrc = (const float4*)(mean1 + (size_t)r0 * DIM);
    for (int i = tid; i < 16 * DIM / 4; i += 256) {
        ((float4*)shA)[i] = gsrc[i];
    }
    if (tid < 16) shSqRow[tid] = sq[r0 + tid];
    __syncthreads();

    const int hl  = lane >> 4;       // 0: lanes 0-15, 1: lanes 16-31
    const int l15 = lane & 15;
    const int kb  = hl * 2;          // A/B K sub-offset per ISA 16x4 f32 layout

    // Private sorted top-5 (ascending squared distance) for lanes 0-15.
    float tv[KNEI]; int ti[KNEI];
#pragma unroll
    for (int k = 0; k < KNEI; ++k) { tv[k] = __builtin_inff(); ti[k] = -1; }

    for (int ct = wave; ct < COLTILES; ct += WAVES) {
        const int c0 = ct * 16;
        v8f acc = {0.f, 0.f, 0.f, 0.f, 0.f, 0.f, 0.f, 0.f};

        // A: row l15 of the staged block; lane half selects K pair {0,1}/{2,3}.
        const float* abase = shA + l15 * DIM + kb;
        // B: column c0+l15 of the Gram = row c0+l15 of mean1 (same K striping).
        const float* bbase = mean1 + (size_t)(c0 + l15) * DIM + kb;

#pragma unroll 8
        for (int kt = 0; kt < DIM / 4; ++kt) {
            v2f a = *(const v2f*)(abase + kt * 4);   // ds_load_b64
            v2f b = *(const v2f*)(bbase + kt * 4);   // global L2-resident load
            acc = __builtin_amdgcn_wmma_f32_16x16x4_f32(
                /*neg_a=*/false, a, /*neg_b=*/false, b,
                /*c_mod=*/(short)0, acc, /*reuse_a=*/false, /*reuse_b=*/false);
        }

        // acc[v] = dot(row r0+M, row c0+N) with M = v + 8*hl, N = l15.
        const float sqc = sq[c0 + l15];
#pragma unroll
        for (int v = 0; v < 8; ++v) {
            const int M = v + hl * 8;
            float d2 = fmaxf(shSqRow[M] + sqc - 2.0f * acc[v], 0.0f);
            if (r0 + M == c0 + l15) d2 = __builtin_inff();   // mask diagonal
            shTile[wave][M][l15] = d2;
        }
        __syncthreads();   // uniform: every wave runs exactly COLTILES/WAVES iters

        if (lane < 16) {   // lane m scans its row of this wave's tile
            const int m = lane;
#pragma unroll
            for (int n = 0; n < 16; ++n) {
                float d = shTile[wave][m][n];
                if (d < tv[KNEI - 1]) {
                    int j = c0 + n;
                    int p = KNEI - 1;
                    while (p > 0 && tv[p - 1] > d) {
                        tv[p] = tv[p - 1]; ti[p] = ti[p - 1]; --p;
                    }
                    tv[p] = d; ti[p] = j;
                }
            }
        }
        __syncthreads();
    }

    // Publish per-wave top-5, then merge 8 lists per row into final top-5.
    if (lane < 16) {
#pragma unroll
        for (int k = 0; k < KNEI; ++k) {
            shVal[lane][wave][k] = tv[k];
            shIdx[lane][wave][k] = ti[k];
        }
    }
    __syncthreads();

    if (tid < 16) {
        float mv[KNEI]; int mi[KNEI];
#pragma unroll
        for (int k = 0; k < KNEI; ++k) { mv[k] = __builtin_inff(); mi[k] = -1; }
        for (int w = 0; w < WAVES; ++w) {
#pragma unroll
            for (int k = 0; k < KNEI; ++k) {
                float d = shVal[tid][w][k];
                if (d < mv[KNEI - 1]) {
                    int j = shIdx[tid][w][k];
                    int p = KNEI - 1;
                    while (p > 0 && mv[p - 1] > d) {
                        mv[p] = mv[p - 1]; mi[p] = mi[p - 1]; --p;
                    }
                    mv[p] = d; mi[p] = j;
                }
            }
        }
#pragma unroll
        for (int k = 0; k < KNEI; ++k)
            topk_idx[(size_t)(r0 + tid) * KNEI + k] = mi[k];
    }
}

// ---------------------------------------------------------------------------
// Kernel 3: per-element loss -> fixed-order per-block partial sums.
// ---------------------------------------------------------------------------
__global__ __launch_bounds__(256)
void ndl_loss_partial(const float* __restrict__ mean1,
                      const float* __restrict__ var1,
                      const int* __restrict__ topk_idx,
                      float* __restrict__ partial) {
    __shared__ float red[256];
    const int tid = threadIdx.x;
    const int total = BATCH * DIM;
    float sum = 0.0f;
    for (int idx = blockIdx.x * 256 + tid; idx < total; idx += gridDim.x * 256) {
        const int i = idx >> 9;          // / DIM
        const int d = idx & (DIM - 1);   // % DIM
        const int* nb = topk_idx + (size_t)i * KNEI;
        float x0 = mean1[(size_t)nb[0] * DIM + d];
        float x1 = mean1[(size_t)nb[1] * DIM + d];
        float x2 = mean1[(size_t)nb[2] * DIM + d];
        float x3 = mean1[(size_t)nb[3] * DIM + d];
        float x4 = mean1[(size_t)nb[4] * DIM + d];
        float mu = (x0 + x1 + x2 + x3 + x4) * 0.2f;
        float e0 = x0 - mu, e1 = x1 - mu, e2 = x2 - mu, e3 = x3 - mu, e4 = x4 - mu;
        float var = (e0 * e0 + e1 * e1 + e2 * e2 + e3 * e3 + e4 * e4) * 0.25f; // ddof=1
        float nstd = __builtin_amdgcn_sqrtf(var);   // raw v_sqrt_f32, ~1ulp
        float pred = expf(0.5f * var1[idx]);        // sqrt(exp(v)) == exp(v/2)
        float err  = pred - nstd;
        sum += err * err;
    }
    red[tid] = sum;
    __syncthreads();
    for (int st = 128; st > 0; st >>= 1) {
        if (tid < st) red[tid] += red[tid + st];
        __syncthreads();
    }
    if (tid == 0) partial[blockIdx.x] = red[0];
}

// ---------------------------------------------------------------------------
// Kernel 4: deterministic final reduction of NPART partials -> scalar loss.
// ---------------------------------------------------------------------------
__global__ __launch_bounds__(256)
void ndl_finalize(const float* __restrict__ partial, float* __restrict__ out) {
    __shared__ float red[256];
    const int tid = threadIdx.x;
    float s = 0.0f;
    for (int i = tid; i < NPART; i += 256) s += partial[i];
    red[tid] = s;
    __syncthreads();
    for (int st = 128; st > 0; st >>= 1) {
        if (tid < st) red[tid] += red[tid + st];
        __syncthreads();
    }
    if (tid == 0) out[0] = red[0] / (float)(BATCH * DIM);
}

// ---------------------------------------------------------------------------
extern "C" void kernel_launch(void* const* d_in, const int* in_sizes, int n_in,
                              void* d_out, int out_size, void* d_ws, size_t ws_size,
                              hipStream_t stream) {
    const float* mean1 = (const float*)d_in[0];
    const float* var1  = (const float*)d_in[1];
    // d_in[2] (mean2) and d_in[3] (var2) are unused per the reference.

    char* ws = (char*)d_ws;
    float* sq      = (float*)ws;                                   // 4096 f32
    int*   topk    = (int*)(ws + (size_t)BATCH * sizeof(float));   // 4096*5 i32
    float* partial = (float*)(ws + (size_t)BATCH * sizeof(float)
                                 + (size_t)BATCH * KNEI * sizeof(int)); // 1024 f32

    ndl_rownorm   <<<BATCH, 128, 0, stream>>>(mean1, sq);
    ndl_dist_topk <<<BATCH / 16, 256, 0, stream>>>(mean1, sq, topk);
    ndl_loss_partial<<<NPART, 256, 0, stream>>>(mean1, var1, topk, partial);
    ndl_finalize  <<<1, 256, 0, stream>>>(partial, (float*)d_out);
}